// __AddDecomposedRelativePositionsTF_74723841016424
// MI455X (gfx1250) — compile-verified
//
#include <hip/hip_runtime.h>
#include <hip/hip_bf16.h>

typedef float v2f __attribute__((ext_vector_type(2)));
typedef float v4f __attribute__((ext_vector_type(4)));
typedef float v8f __attribute__((ext_vector_type(8)));

// Problem constants (fixed by setup_inputs): B=12, QH=QW=KH=KW=64, C=64, L=127
#define NB   12
#define S    64
#define NC   64
#define REL_ELEMS (NB * S * S * S)   // 12*64*64*64 = 3,145,728 floats per rel tensor

// ---------------------------------------------------------------------------
// Stage 1: compute rel_h and rel_w with V_WMMA_F32_16X16X4_F32 (full f32).
//   rel_h[b,h,w,k] = sum_c q[b, h*64+w, c] * rel_pos_h[(h-k+63), c]
//   rel_w[b,h,w,k] = sum_c q[b, h*64+w, c] * rel_pos_w[(w-k+63), c]
// One wave computes one 16x16 output tile, K-loop of 16 steps (K=4 each).
// Grid: 2(sel) * 64(fixedIdx) * 48(m-tiles over r=b*64+x) * 4(n-tiles) waves.
// ---------------------------------------------------------------------------
__global__ __launch_bounds__(256) void relpos_gemm_wmma(
    const float* __restrict__ q,          // (12, 4096, 64)
    const float* __restrict__ rel_pos_h,  // (127, 64)
    const float* __restrict__ rel_pos_w,  // (127, 64)
    float* __restrict__ relh_out,         // (12, 64, 64, 64) [b][h][w][k]
    float* __restrict__ relw_out)         // (12, 64, 64, 64) [b][h][w][k]
{
    const int wid  = blockIdx.x * 8 + (threadIdx.x >> 5);
    const int lane = threadIdx.x & 31;

    const int n0       = wid & 3;            // 0..3   (k-tile)
    const int m0       = (wid >> 2) % 48;    // 0..47  (r-tile, r = b*64 + x)
    const int fixedIdx = (wid / 192) & 63;   // h (sel=0) or w (sel=1)
    const int sel      = wid / 12288;        // 0 = rel_h, 1 = rel_w

    const int lm = lane & 15;                // M (for A) / N (for B,D)
    const int g  = lane >> 4;                // half selector

    // ---- A fragment addressing: A[M, K] = q[row(r), c], r = m0*16 + lm ----
    const int r = m0 * 16 + lm;
    const int b = r >> 6;
    const int x = r & 63;
    const long qrow = sel ? ((long)b * 4096 + (long)x * 64 + fixedIdx)
                          : ((long)b * 4096 + (long)fixedIdx * 64 + x);
    const float* aptr = q + qrow * NC + 2 * g;   // lane holds K = 2g, 2g+1

    // ---- B fragment addressing: B[K, N] = rel_pos[idx(n), c], n = n0*16+lm ----
    const int n    = n0 * 16 + lm;
    const int ridx = fixedIdx - n + 63;          // always in [0, 126]
    const float* rp   = sel ? rel_pos_w : rel_pos_h;
    const float* bptr = rp + (long)ridx * NC + 2 * g;

    v8f acc = {};
#pragma unroll
    for (int kk = 0; kk < 16; ++kk) {
        v2f a  = *(const v2f*)(aptr + kk * 4);
        v2f bb = *(const v2f*)(bptr + kk * 4);
        // D = A(16x4, f32) x B(4x16, f32) + C ; full f32 accumulate
        acc = __builtin_amdgcn_wmma_f32_16x16x4_f32(
            /*neg_a=*/false, a, /*neg_b=*/false, bb,
            /*c_mod=*/(short)0, acc, /*reuse_a=*/false, /*reuse_b=*/false);
    }

    // ---- D store: lane L, vgpr v -> row = m0*16 + v + 8*g, col k = n ----
    float* outp = sel ? relw_out : relh_out;
#pragma unroll
    for (int v = 0; v < 8; ++v) {
        const int row = m0 * 16 + v + 8 * g;
        const int ob  = row >> 6;
        const int ox  = row & 63;
        const long flat = sel
            ? ((long)ob * 262144 + (long)ox * 4096 + (long)fixedIdx * 64 + n)
            : ((long)ob * 262144 + (long)fixedIdx * 4096 + (long)ox * 64 + n);
        outp[flat] = acc[v];
    }
}

// ---------------------------------------------------------------------------
// Stage 2: pure-bandwidth broadcast add (1.61 GB stream, floor ~69us @23.3TB/s)
//   out[b,h,w,kh,kw] = attn[...] + rel_h[b,h,w,kh] + rel_w[b,h,w,kw]
// One block per (b,h,w) slab of 4096 floats; rel vectors staged in LDS;
// non-temporal hints on the big stream (read-once / write-once).
// ---------------------------------------------------------------------------
__global__ __launch_bounds__(256) void add_decomposed_bias(
    const float* __restrict__ attn,   // (12, 4096, 4096)
    const float* __restrict__ relh,   // (12*64*64, 64)
    const float* __restrict__ relw,   // (12*64*64, 64)
    float* __restrict__ out)          // (12, 4096, 4096)
{
    __shared__ __align__(16) float sh[64];
    __shared__ __align__(16) float sw[64];

    const long bhw = blockIdx.x;          // 0 .. 49151
    const int  t   = threadIdx.x;

    if (t < 64)        sw[t]      = relw[bhw * 64 + t];
    else if (t < 128)  sh[t - 64] = relh[bhw * 64 + (t - 64)];
    __syncthreads();

    const v4f* in4  = (const v4f*)(attn + bhw * 4096);
    v4f*       out4 = (v4f*)(out + bhw * 4096);

#pragma unroll
    for (int j = 0; j < 4; ++j) {
        const int i4 = t + j * 256;        // float4 index 0..1023
        const int e  = i4 << 2;            // element index, multiple of 4
        const int kh = e >> 6;
        const int kw = e & 63;             // multiple of 4 -> aligned LDS v4f

        v4f v = __builtin_nontemporal_load(&in4[i4]);
        const float rh = sh[kh];
        const v4f   w4 = *(const v4f*)(&sw[kw]);
        v += rh + w4;                      // vector + splat-scalar + vector
        __builtin_nontemporal_store(v, &out4[i4]);
    }
}

extern "C" void kernel_launch(void* const* d_in, const int* in_sizes, int n_in,
                              void* d_out, int out_size, void* d_ws, size_t ws_size,
                              hipStream_t stream) {
    const float* attn      = (const float*)d_in[0];  // (12, 4096, 4096)
    const float* q         = (const float*)d_in[1];  // (12, 4096, 64)
    const float* rel_pos_h = (const float*)d_in[2];  // (127, 64)
    const float* rel_pos_w = (const float*)d_in[3];  // (127, 64)
    // d_in[4..7] are the q_h/q_w/k_h/k_w scalars (fixed = 64), unused.

    float* relh = (float*)d_ws;                      // 12.6 MB
    float* relw = relh + REL_ELEMS;                  // 12.6 MB (total 25.2 MB)
    float* out  = (float*)d_out;

    // Stage 1: 24576 tile-waves / 8 waves per block = 3072 blocks
    relpos_gemm_wmma<<<3072, 256, 0, stream>>>(q, rel_pos_h, rel_pos_w, relh, relw);

    // Stage 2: one block per (b,h,w) slab
    add_decomposed_bias<<<NB * S * S, 256, 0, stream>>>(attn, relh, relw, out);
}